// ContrastiveLoss_66632122630347
// MI455X (gfx1250) — compile-verified
//
#include <hip/hip_runtime.h>
#include <math.h>

// ---------------------------------------------------------------------------
// Fused contrastive loss for gfx1250 (MI455X), f32 WMMA path.
//   sim = exp((X @ X^T) * TEMP); per-row neg_sum/max_neg; loss + accuracy.
// Two fused GEMM passes, no materialized N x N matrix.
// Round 2: double-buffered B chunks (clause loads, hidden L2 latency),
//          dual WMMA accumulator chains, branchless epilogue with __expf.
// ---------------------------------------------------------------------------

typedef __attribute__((ext_vector_type(2))) float v2f;
typedef __attribute__((ext_vector_type(8))) float v8f;

#define TEMP_   0.02f
#define OTHERW_ 0.5f
#define FDIM    512      // feature dim (from reference)
#define LDSS    516      // padded LDS row stride (516 % 64 == 4 -> no bank conflicts)

__device__ __forceinline__ const float* vrow(const float* f1, const float* f2,
                                             int r, int n1) {
  return (r < n1) ? (f1 + (size_t)r * FDIM) : (f2 + (size_t)(r - n1) * FDIM);
}

// order-preserving float <-> uint mapping for atomic max
__device__ __forceinline__ unsigned fflip(float f) {
  unsigned u = __float_as_uint(f);
  return (u & 0x80000000u) ? ~u : (u | 0x80000000u);
}
__device__ __forceinline__ float funflip(unsigned u) {
  return (u & 0x80000000u) ? __uint_as_float(u & 0x7fffffffu)
                           : __uint_as_float(~u);
}

// Stage the 16 x 512 A-band for row-tile I into LDS (float4 cooperative copy).
__device__ __forceinline__ void load_atile(float* As, const float* f1,
                                           const float* f2, int I, int n1) {
  for (int idx = threadIdx.x; idx < 16 * (FDIM / 4); idx += blockDim.x) {
    int r  = idx >> 7;             // FDIM/4 == 128
    int c4 = (idx & 127) << 2;
    const float* src = vrow(f1, f2, I * 16 + r, n1) + c4;
    *(float4*)(As + r * LDSS + c4) = *(const float4*)src;
  }
}

// 16x16 f32 dot tile via 128 x v_wmma_f32_16x16x4_f32.
// Fragment layout (ISA 7.12.2): lane l supplies row (l&15), k-pair 2*(l>>4)+{0,1}.
// B is software-pipelined in 32-K chunks (8 global_load_b64 issued as a clause
// one chunk ahead); two accumulator chains hide WMMA accumulate latency.
__device__ __forceinline__ v8f tile_dot(const float* aLds, const float* brow,
                                        int hi) {
  const int base = hi << 1;
  v8f c0 = {0.f, 0.f, 0.f, 0.f, 0.f, 0.f, 0.f, 0.f};
  v8f c1 = {0.f, 0.f, 0.f, 0.f, 0.f, 0.f, 0.f, 0.f};

  v2f b[8];
#pragma unroll
  for (int i = 0; i < 8; ++i)
    b[i] = *(const v2f*)(brow + base + (i << 2));

#pragma unroll
  for (int ch = 0; ch < FDIM / 32; ++ch) {
    const float* aPtr = aLds + (ch << 5) + base;
    v2f bn[8];
    if (ch < FDIM / 32 - 1) {
#pragma unroll
      for (int i = 0; i < 8; ++i)
        bn[i] = *(const v2f*)(brow + ((ch + 1) << 5) + base + (i << 2));
    }
#pragma unroll
    for (int i = 0; i < 8; ++i) {
      v2f a = *(const v2f*)(aPtr + (i << 2));   // ds_load_b64, conflict-free
      if (i & 1)
        c1 = __builtin_amdgcn_wmma_f32_16x16x4_f32(false, a, false, b[i],
                                                   (short)0, c1, false, false);
      else
        c0 = __builtin_amdgcn_wmma_f32_16x16x4_f32(false, a, false, b[i],
                                                   (short)0, c0, false, false);
    }
    if (ch < FDIM / 32 - 1) {
#pragma unroll
      for (int i = 0; i < 8; ++i) b[i] = bn[i];
    }
  }
  return c0 + c1;
}

// ---------------------------------------------------------------------------
__global__ void cl_init(const int* overlap, int* labels, float* negsum,
                        unsigned* negmaxb, float* poscw, float* posdot,
                        unsigned* poscnt, float* accum, unsigned* correct,
                        int n1, int N, int K, int bs) {
  int i = blockIdx.x * blockDim.x + threadIdx.x;
  if (i == 0) { accum[0] = 0.f; accum[1] = 0.f; *correct = 0u; }
  if (i >= N) return;
  int lab;
  if (i < n1) {
    lab = i / bs;
  } else {
    int g = (i - n1) / bs;
    if (overlap[g] != 0) lab = g;
    else {
      int e = 0;
      for (int j = 0; j < g; ++j) e += (overlap[j] == 0);
      lab = K + e;
    }
  }
  labels[i] = lab;
  negsum[i] = 0.f; negmaxb[i] = 0u; poscw[i] = 0.f; posdot[i] = 0.f;
  poscnt[i] = 0u;
}

// Pass 1: per-row neg_sum, max_neg(dot), sum of cross-weights over positives,
//         and sum of cross * (-dot*TEMP) over positives.
__global__ __launch_bounds__(256) void cl_pass1(
    const float* f1, const float* f2, const int* labels, float* negsum,
    unsigned* negmaxb, float* poscw, float* posdot, unsigned* poscnt, int n1,
    int N) {
  __shared__ float As[16 * LDSS];
  int I = blockIdx.x;
  load_atile(As, f1, f2, I, n1);
  __syncthreads();

  int lane = threadIdx.x & 31, wv = threadIdx.x >> 5;
  int l15 = lane & 15, hi = lane >> 4;
  int rowbase = I * 16 + (hi << 3);

  int rlab[8], rh[8];
#pragma unroll
  for (int v = 0; v < 8; ++v) {
    rlab[v] = labels[rowbase + v];
    rh[v]   = (rowbase + v) < n1;
  }
  float aNS[8], aNM[8], aPC[8], aPD[8];
  int aCN[8];
#pragma unroll
  for (int v = 0; v < 8; ++v) {
    aNS[v] = 0.f; aNM[v] = -INFINITY; aPC[v] = 0.f; aPD[v] = 0.f; aCN[v] = 0;
  }

  const float* aLds = As + l15 * LDSS;
  int nT = N >> 4;
  for (int J = wv; J < nT; J += 8) {
    const float* brow = vrow(f1, f2, J * 16 + l15, n1);
    v8f c = tile_dot(aLds, brow, hi);

    int col  = J * 16 + l15;
    int clab = labels[col];
    int colh = col < n1;
    float cf[8];
#pragma unroll
    for (int v = 0; v < 8; ++v) cf[v] = c[v];
#pragma unroll
    for (int v = 0; v < 8; ++v) {               // fully branchless (v_cndmask)
      int row   = rowbase + v;
      float d   = cf[v];
      bool same = (rlab[v] == clab);
      bool pos  = same && (row != col);
      float e   = __expf(d * TEMP_);            // native v_exp_f32 path
      aNS[v] += same ? 0.f : e;
      aNM[v]  = same ? aNM[v] : fmaxf(aNM[v], d);
      float cw = (rh[v] == colh) ? 1.0f : OTHERW_;
      float pw = pos ? cw : 0.f;
      aPC[v] += pw;
      aPD[v] -= pw * (d * TEMP_);
      aCN[v] += pos ? 1 : 0;
    }
  }

  // reduce across the 16 column-lanes of each half-wave (same rows per half)
#pragma unroll
  for (int v = 0; v < 8; ++v) {
    for (int m = 1; m < 16; m <<= 1) {
      aNS[v] += __shfl_xor(aNS[v], m);
      aNM[v]  = fmaxf(aNM[v], __shfl_xor(aNM[v], m));
      aPC[v] += __shfl_xor(aPC[v], m);
      aPD[v] += __shfl_xor(aPD[v], m);
      aCN[v] += __shfl_xor(aCN[v], m);
    }
  }
  if (l15 == 0) {
#pragma unroll
    for (int v = 0; v < 8; ++v) {
      int row = rowbase + v;
      atomicAdd(&negsum[row], aNS[v]);
      atomicMax(&negmaxb[row], fflip(aNM[v]));
      atomicAdd(&poscw[row], aPC[v]);
      atomicAdd(&posdot[row], aPD[v]);
      atomicAdd(&poscnt[row], (unsigned)aCN[v]);
    }
  }
}

// loss numerator = sum_r [ posdot[r] + poscw[r]*log(negsum[r]) ]; total_pos.
__global__ void cl_finalize(const float* negsum, const float* poscw,
                            const float* posdot, const unsigned* poscnt,
                            float* accum, int N) {
  __shared__ float sNum[256];
  __shared__ float sPos[256];
  float ln = 0.f, lp = 0.f;
  for (int r = threadIdx.x; r < N; r += 256) {
    ln += posdot[r] + poscw[r] * logf(negsum[r]);
    lp += (float)poscnt[r];
  }
  sNum[threadIdx.x] = ln;
  sPos[threadIdx.x] = lp;
  __syncthreads();
  for (int s = 128; s > 0; s >>= 1) {
    if (threadIdx.x < (unsigned)s) {
      sNum[threadIdx.x] += sNum[threadIdx.x + s];
      sPos[threadIdx.x] += sPos[threadIdx.x + s];
    }
    __syncthreads();
  }
  if (threadIdx.x == 0) { accum[0] = sNum[0]; accum[1] = sPos[0]; }
}

// Pass 2: recompute dots, count positives beating the row's max negative.
__global__ __launch_bounds__(256) void cl_pass2(const float* f1,
                                                const float* f2,
                                                const int* labels,
                                                const unsigned* negmaxb,
                                                unsigned* correct, int n1,
                                                int N) {
  __shared__ float As[16 * LDSS];
  int I = blockIdx.x;
  load_atile(As, f1, f2, I, n1);
  __syncthreads();

  int lane = threadIdx.x & 31, wv = threadIdx.x >> 5;
  int l15 = lane & 15, hi = lane >> 4;
  int rowbase = I * 16 + (hi << 3);

  int rlab[8];
  float nm[8];
#pragma unroll
  for (int v = 0; v < 8; ++v) {
    rlab[v] = labels[rowbase + v];
    nm[v]   = funflip(negmaxb[rowbase + v]);
  }

  const float* aLds = As + l15 * LDSS;
  int nT = N >> 4;
  int corr = 0;
  for (int J = wv; J < nT; J += 8) {
    const float* brow = vrow(f1, f2, J * 16 + l15, n1);
    v8f c = tile_dot(aLds, brow, hi);
    int col  = J * 16 + l15;
    int clab = labels[col];
    float cf[8];
#pragma unroll
    for (int v = 0; v < 8; ++v) cf[v] = c[v];
#pragma unroll
    for (int v = 0; v < 8; ++v) {
      int row  = rowbase + v;
      bool pos = (rlab[v] == clab) && (row != col);
      corr += (pos && (cf[v] > nm[v])) ? 1 : 0;
    }
  }
  for (int m = 1; m < 32; m <<= 1) corr += __shfl_xor(corr, m);
  if (lane == 0) atomicAdd(correct, (unsigned)corr);
}

__global__ void cl_write(const float* accum, const unsigned* correct,
                         float* out) {
  if (threadIdx.x == 0 && blockIdx.x == 0) {
    float tp = accum[1];
    out[0] = (float)(*correct) / tp;   // accuracy
    out[1] = accum[0] / tp;            // loss
  }
}

// ---------------------------------------------------------------------------
extern "C" void kernel_launch(void* const* d_in, const int* in_sizes, int n_in,
                              void* d_out, int out_size, void* d_ws,
                              size_t ws_size, hipStream_t stream) {
  (void)n_in; (void)out_size; (void)ws_size;
  const float* f1      = (const float*)d_in[0];
  const float* f2      = (const float*)d_in[1];
  const int*   overlap = (const int*)d_in[2];
  // bs lives on device (d_in[3]); recover it on host from sizes instead.
  int K  = in_sizes[2];
  int n1 = in_sizes[0] / FDIM;
  int n2 = in_sizes[1] / FDIM;
  int N  = n1 + n2;
  int bs = n1 / K;

  char* p = (char*)d_ws;
  int*      labels  = (int*)p;      p += (size_t)N * 4;
  float*    negsum  = (float*)p;    p += (size_t)N * 4;
  unsigned* negmaxb = (unsigned*)p; p += (size_t)N * 4;
  float*    poscw   = (float*)p;    p += (size_t)N * 4;
  float*    posdot  = (float*)p;    p += (size_t)N * 4;
  unsigned* poscnt  = (unsigned*)p; p += (size_t)N * 4;
  float*    accum   = (float*)p;    p += 16;
  unsigned* correct = (unsigned*)p;
  float*    out     = (float*)d_out;

  int nT = N >> 4;
  int initBlocks = (N + 255) / 256;
  cl_init<<<initBlocks, 256, 0, stream>>>(overlap, labels, negsum, negmaxb,
                                          poscw, posdot, poscnt, accum,
                                          correct, n1, N, K, bs);
  cl_pass1<<<nT, 256, 0, stream>>>(f1, f2, labels, negsum, negmaxb, poscw,
                                   posdot, poscnt, n1, N);
  cl_finalize<<<1, 256, 0, stream>>>(negsum, poscw, posdot, poscnt, accum, N);
  cl_pass2<<<nT, 256, 0, stream>>>(f1, f2, labels, negmaxb, correct, n1, N);
  cl_write<<<1, 1, 0, stream>>>(accum, correct, out);
}